// Standard_83476984365425
// MI455X (gfx1250) — compile-verified
//
#include <hip/hip_runtime.h>
#include <hip/hip_bf16.h>

typedef __attribute__((ext_vector_type(2))) float v2f;
typedef __attribute__((ext_vector_type(4))) float v4f;
typedef __attribute__((ext_vector_type(8))) float v8f;

#define LDS_STRIDE 20          // words per 16-float row (16B-aligned, bank-spread)
#define WAVE_LDS   640         // floats per wave: X tile (320) + U^T tile (320)

// Block-wise 256-pt Hadamard: y_block = (H16 (x) H16)/16 * x_block
// => reshape block to 16x16 X (row-major), Y = (1/16) * H16 * X * H16.
// Each 16x16x16 matmul = 4x V_WMMA_F32_16X16X4_F32 (exact fp32, H entries +-1).
__global__ __launch_bounds__(256) void hadamard256_wmma(
    const float* __restrict__ x, float* __restrict__ out,
    int nblocks, int nwaves)
{
    __shared__ float lds[8 * WAVE_LDS];
    const int lane  = threadIdx.x & 31;
    const int wslot = threadIdx.x >> 5;            // wave within workgroup
    const int wid   = blockIdx.x * 8 + wslot;      // global wave id
    float* Xs = lds + wslot * WAVE_LDS;            // 16x16 X, row stride 20
    float* Ut = Xs + 320;                          // U transposed (col-major)

    const int h  = lane >> 4;                      // lane half: 0/1
    const int mn = lane & 15;                      // A-row m == B-col n for this lane

    // H16[i][j] = (-1)^popc(i&j), symmetric. Per-lane constant fragments:
    //   bH[s]: B-frag for U = X*H16   (B[k][n] = H16[k][n], k = 4s+2h+v)
    //   aH[s]: A-frag for Y = H16*U, with the 1/16 normalization folded in
    v2f bH[4], aH[4];
#pragma unroll
    for (int s = 0; s < 4; ++s) {
#pragma unroll
        for (int v = 0; v < 2; ++v) {
            int k = 4 * s + 2 * h + v;
            float sg = (__popc(k & mn) & 1) ? -1.0f : 1.0f;
            bH[s][v] = sg;
            aH[s][v] = sg * 0.0625f;               // exact power-of-two scale
        }
    }

    const int row0 = lane >> 2;                    // staging row (0..7)
    const int col0 = (lane & 3) * 4;               // staging col (0,4,8,12)

    for (int blk = wid; blk < nblocks; blk += nwaves) {   // wave-uniform => EXEC all-1
        const float* src = x + (size_t)blk * 256;
        float*       dst = out + (size_t)blk * 256;

        // ---- stage X (16x16 row-major) into LDS, coalesced b128 ----
        v4f r0 = *(const v4f*)(src + lane * 4);
        v4f r1 = *(const v4f*)(src + 128 + lane * 4);
        *(v4f*)(Xs + row0 * LDS_STRIDE + col0)       = r0;
        *(v4f*)(Xs + (row0 + 8) * LDS_STRIDE + col0) = r1;

        // ---- U = X * H16 : A from LDS, B = +-1 constants in regs ----
        v8f u = {};
#pragma unroll
        for (int s = 0; s < 4; ++s) {
            v2f aX = *(const v2f*)(Xs + mn * LDS_STRIDE + 4 * s + 2 * h);
            u = __builtin_amdgcn_wmma_f32_16x16x4_f32(false, aX, false, bH[s],
                                                      (short)0, u, false, false);
        }

        // ---- store U transposed: Ut[n*20 + row]; lane holds U[r+8h][n] in u[r] ----
        *(v4f*)(Ut + mn * LDS_STRIDE + 8 * h)     = *(v4f*)&u;
        *(v4f*)(Ut + mn * LDS_STRIDE + 8 * h + 4) = *((v4f*)&u + 1);

        // ---- Y = (H16/16) * U : A = scaled +-1 constants, B from LDS ----
        v8f y = {};
#pragma unroll
        for (int s = 0; s < 4; ++s) {
            v2f bU = *(const v2f*)(Ut + mn * LDS_STRIDE + 4 * s + 2 * h);
            y = __builtin_amdgcn_wmma_f32_16x16x4_f32(false, aH[s], false, bU,
                                                      (short)0, y, false, false);
        }

        // ---- write out: lane holds Y[r+8h][n]; each store = 2x64B dense runs ----
#pragma unroll
        for (int r = 0; r < 8; ++r)
            dst[(r + 8 * h) * 16 + mn] = y[r];
    }
}

extern "C" void kernel_launch(void* const* d_in, const int* in_sizes, int n_in,
                              void* d_out, int out_size, void* d_ws, size_t ws_size,
                              hipStream_t stream) {
    const float* x = (const float*)d_in[0];
    // d_in[1] (H) is unused: Sylvester signs are synthesized via popcount.
    float* out = (float*)d_out;

    int n       = in_sizes[0];
    int nblocks = n / 256;                 // 262144 for the reference shapes

    int wgs = 4096;                        // 32768 waves, 8 blocks per wave
    if (nblocks < wgs * 8) wgs = (nblocks + 7) / 8;
    if (wgs < 1) wgs = 1;
    int nwaves = wgs * 8;

    hadamard256_wmma<<<wgs, 256, 0, stream>>>(x, out, nblocks, nwaves);
}